// fusionLossFunc_improved_87686052316037
// MI455X (gfx1250) — compile-verified
//
#include <hip/hip_runtime.h>
#include <hip/hip_bf16.h>

// Problem constants (from reference setup_inputs)
#define Bb   2
#define Ll   19
#define Cc   (3 * Ll)      // 57
#define Hh   800
#define Ww   640
#define NPIX (Hh * Ww)     // 512000
#define R1SQ 1681.0f       // 41^2
#define R2f  100.0f

#define NBLK 64            // blocks per logits channel in the streaming kernel
#define TPB  256           // 8 waves per block (wave32)

typedef __attribute__((ext_vector_type(2))) float v2f;
typedef __attribute__((ext_vector_type(8))) float v8f;

#if defined(__gfx1250__) && __has_builtin(__builtin_amdgcn_wmma_f32_16x16x4_f32)
#define USE_WMMA_REDUCE 1
#else
#define USE_WMMA_REDUCE 0
#endif

// ---------------------------------------------------------------------------
// Wave32 sum-reduction.
// WMMA path: A is the 16x4 f32 matrix (2 VGPRs/lane). Per the CDNA5 layout,
// lane m (m<16) holds A[m][0],A[m][1]; lane m+16 holds A[m][2],A[m][3].
// Putting p in comp0 and 0 in comp1, with B = all-ones (4x16), gives
// D[m][n] = p_m + p_{m+16}. D VGPR r holds rows r (lanes 0-15) / r+8
// (lanes 16-31), so per-lane sum of the 8 accumulators gives half-wave row
// totals; one xor-16 shuffle completes the full wave sum in exact f32.
// ---------------------------------------------------------------------------
__device__ __forceinline__ float waveReduceSum(float p) {
#if USE_WMMA_REDUCE
    v2f a;
    a.x = p;
    a.y = 0.0f;
    v2f b;
    b.x = 1.0f;
    b.y = 1.0f;
    v8f c = {0.0f, 0.0f, 0.0f, 0.0f, 0.0f, 0.0f, 0.0f, 0.0f};
    c = __builtin_amdgcn_wmma_f32_16x16x4_f32(
        /*neg_a=*/false, a, /*neg_b=*/false, b,
        /*c_mod=*/(short)0, c, /*reuse_a=*/false, /*reuse_b=*/false);
    float s = ((c[0] + c[1]) + (c[2] + c[3])) + ((c[4] + c[5]) + (c[6] + c[7]));
    s += __shfl_xor(s, 16, 32);
    return s; // all lanes hold the wave total
#else
    for (int off = 16; off > 0; off >>= 1) p += __shfl_xor(p, off, 32);
    return p;
#endif
}

// Deterministic block reduction; result valid on threadIdx.x == 0.
// `wsum` is shared memory with >= blockDim.x/32 floats.
__device__ __forceinline__ float blockReduceSum(float p, float* wsum) {
    float t = waveReduceSum(p);
    const int wid  = threadIdx.x >> 5;
    const int lane = threadIdx.x & 31;
    __syncthreads();
    if (lane == 0) wsum[wid] = t;
    __syncthreads();
    float r = 0.0f;
    if (threadIdx.x == 0) {
        const int nw = blockDim.x >> 5;
        for (int i = 0; i < nw; ++i) r += wsum[i]; // serial: deterministic
    }
    return r;
}

__device__ __forceinline__ float softplus_f(float z) {
    // max(z,0) + log(1 + exp(-|z|)) == softplus(z), numerically stable
    return fmaxf(z, 0.0f) + __logf(1.0f + __expf(-fabsf(z)));
}

// ---------------------------------------------------------------------------
// Kernel A: stream all logit channels, per-block partial of sum(softplus(z)).
// Grid = Bb*Ll*NBLK blocks. partials[chan*NBLK + sub] written uniquely.
// ---------------------------------------------------------------------------
__global__ void bce_stream_kernel(const float* __restrict__ fm,
                                  float* __restrict__ partials) {
    __shared__ float wsum[TPB / 32];
    const int chan = blockIdx.x / NBLK;      // 0 .. Bb*Ll-1
    const int sub  = blockIdx.x % NBLK;
    const int bi   = chan / Ll;
    const int li   = chan % Ll;

    const float4* __restrict__ src =
        (const float4*)(fm + (size_t)(bi * Cc + li) * NPIX);
    const int n4 = NPIX / 4; // 128000 float4 per channel

    float acc = 0.0f;
    for (int idx = sub * TPB + threadIdx.x; idx < n4; idx += NBLK * TPB) {
        float4 v = src[idx];
        acc += softplus_f(v.x);
        acc += softplus_f(v.y);
        acc += softplus_f(v.z);
        acc += softplus_f(v.w);
    }

    float s = blockReduceSum(acc, wsum);
    if (threadIdx.x == 0) partials[blockIdx.x] = s;
}

// ---------------------------------------------------------------------------
// Kernel B: per-(b,l) disc statistics over the clipped 83x83 bounding box.
// cres[chan] = { zsum, cnt, l1x_sum, l1y_sum } (float4).
// ---------------------------------------------------------------------------
__global__ void circle_kernel(const float* __restrict__ fm,
                              const float* __restrict__ lms,
                              float4* __restrict__ cres) {
    __shared__ float wsum[TPB / 32];
    const int chan = blockIdx.x; // 0 .. Bb*Ll-1
    const int bi   = chan / Ll;
    const int li   = chan % Ll;

    const float X = rintf(lms[chan * 2 + 0] * (float)(Hh - 1)); // round-half-even
    const float Y = rintf(lms[chan * 2 + 1] * (float)(Ww - 1));
    const int xi = (int)X;
    const int yi = (int)Y;
    const int i0 = max(0, xi - 41), i1 = min(Hh - 1, xi + 41);
    const int j0 = max(0, yi - 41), j1 = min(Ww - 1, yi + 41);
    const int bh = i1 - i0 + 1;
    const int bw = j1 - j0 + 1;

    const float* __restrict__ lg = fm + (size_t)(bi * Cc + li) * NPIX;
    const float* __restrict__ px = fm + (size_t)(bi * Cc + Ll + li) * NPIX;
    const float* __restrict__ py = fm + (size_t)(bi * Cc + 2 * Ll + li) * NPIX;

    float cnt = 0.0f, zsum = 0.0f, l1x = 0.0f, l1y = 0.0f;
    const int npts = bh * bw;
    for (int p = threadIdx.x; p < npts; p += TPB) {
        const int i = i0 + p / bw;
        const int j = j0 + p % bw;
        const float dX = X - (float)i;
        const float dY = Y - (float)j;
        if (dX * dX + dY * dY <= R1SQ) {
            const size_t off = (size_t)i * Ww + j;
            cnt  += 1.0f;
            zsum += lg[off];
            l1x  += fabsf(px[off] - dX / R2f);
            l1y  += fabsf(py[off] - dY / R2f);
        }
    }

    const float tz = blockReduceSum(zsum, wsum);
    const float tc = blockReduceSum(cnt,  wsum);
    const float tx = blockReduceSum(l1x,  wsum);
    const float ty = blockReduceSum(l1y,  wsum);
    if (threadIdx.x == 0) {
        float4 r;
        r.x = tz; r.y = tc; r.z = tx; r.w = ty;
        cres[chan] = r;
    }
}

// ---------------------------------------------------------------------------
// Finalize: combine per-channel sums into the scalar mean loss.
// ---------------------------------------------------------------------------
__global__ void finalize_kernel(const float* __restrict__ partials,
                                const float4* __restrict__ cres,
                                float* __restrict__ out) {
    __shared__ float terms[Bb * Ll];
    const int t = threadIdx.x;
    if (t < Bb * Ll) {
        float fs = 0.0f;
        for (int k = 0; k < NBLK; ++k) fs += partials[t * NBLK + k];
        const float4 r = cres[t];
        const float zsum = r.x, cnt = r.y, l1x = r.z, l1y = r.w;
        const float bce = (fs - zsum) / (float)NPIX; // mean over pixels
        terms[t] = 2.0f * bce + l1x / cnt + l1y / cnt;
    }
    __syncthreads();
    if (t == 0) {
        float s = 0.0f;
        for (int k = 0; k < Bb * Ll; ++k) s += terms[k];
        out[0] = s / (float)(Bb * Ll);
    }
}

extern "C" void kernel_launch(void* const* d_in, const int* in_sizes, int n_in,
                              void* d_out, int out_size, void* d_ws, size_t ws_size,
                              hipStream_t stream) {
    (void)in_sizes; (void)n_in; (void)out_size; (void)ws_size;
    const float* fm  = (const float*)d_in[0]; // (2, 57, 800, 640) f32
    const float* lms = (const float*)d_in[1]; // (2, 19, 2) f32
    float* out = (float*)d_out;

    // Workspace layout (all slots written each call before being read):
    //   [0 .. Bb*Ll*NBLK)          : per-block softplus partials (2432 floats)
    //   [Bb*Ll*NBLK .. +Bb*Ll*4)   : per-channel {zsum,cnt,l1x,l1y} float4
    float*  partials = (float*)d_ws;
    float4* cres     = (float4*)(partials + Bb * Ll * NBLK); // 16B-aligned

    bce_stream_kernel<<<Bb * Ll * NBLK, TPB, 0, stream>>>(fm, partials);
    circle_kernel<<<Bb * Ll, TPB, 0, stream>>>(fm, lms, cres);
    finalize_kernel<<<1, 64, 0, stream>>>(partials, cres, out);
}